// MotivicIntegrationSystem_27041114096432
// MI455X (gfx1250) — compile-verified
//
#include <hip/hip_runtime.h>

// ---------------------------------------------------------------------------
// Geodesic integration on a learned Riemannian metric (CDNA5 / gfx1250).
// B=8192 elements, d=16, hid=64, 100 serial steps.
// Strategy: one wave owns 16 batch elements end-to-end; the three dominant
// K=64/K=256 contractions run on V_WMMA_F32_16X16X4_F32 with batch rows as M.
// Weights (A: 64KB, W, b) are staged once into padded LDS; per-wave LDS
// buffers handle C-layout <-> A-operand relayout and per-element 16x16 work.
// ---------------------------------------------------------------------------

typedef __attribute__((ext_vector_type(2))) float v2f;
typedef __attribute__((ext_vector_type(4))) float v4f;
typedef __attribute__((ext_vector_type(8))) float v8f;

#define STEP_SIZE 0.01f
#define NEUM_ITERS 12

// ---- LDS layout (float offsets) ----
#define A_PITCH   68                         // 256 x 68 (pad: banks 4c+p)
#define W_PITCH   17                         // 64 x 17
#define A_OFF     0
#define W_OFF     (256 * A_PITCH)            // 17408
#define B_OFF     (W_OFF + 64 * W_PITCH)     // 18496
#define WV_OFF    (B_OFF + 64)               // 18560
#define ABUF_PITCH 68                        // 16 el x 68
#define M_EPITCH  276                        // element pitch for 16x16 mats
#define M_RPITCH  17                         // row pitch inside an element
#define PW_ABUF   0
#define PW_MBUF   (16 * ABUF_PITCH)          // 1088
#define PW_NBUF   (PW_MBUF + 16 * M_EPITCH)  // 5504
#define PW_SIZE   (PW_NBUF + 16 * M_EPITCH)  // 9920 floats / wave
#define LDS_FLOATS (WV_OFF + 4 * PW_SIZE)    // 58240 floats = 232,960 B

__device__ __forceinline__ v8f wmma4(v2f a, v2f b, v8f c) {
  // D = A(16x4 f32) * B(4x16 f32) + C(16x16 f32)
  return __builtin_amdgcn_wmma_f32_16x16x4_f32(false, a, false, b, (short)0, c,
                                               false, false);
}

__device__ __forceinline__ float fast_tanh(float y) {
  float e2 = __expf(2.0f * y);
  return __fdividef(e2 - 1.0f, e2 + 1.0f);
}

__global__ void __launch_bounds__(128, 1)
geodesic_wmma_kernel(const float* __restrict__ X0, const float* __restrict__ V0,
                     const float* __restrict__ Wg, const float* __restrict__ bg,
                     const float* __restrict__ Ag, const int* __restrict__ stepsPtr,
                     float* __restrict__ out, int Btot) {
  extern __shared__ __align__(16) float smem[];
  const int tid  = threadIdx.x;
  const int lane = tid & 31;
  const int wave = tid >> 5;
  const int half = lane >> 4;     // 0: K lanes 0..1, 1: K lanes 2..3 of each k4
  const int l16  = lane & 15;
  const int steps = *stepsPtr;

  // ---- cooperative weight staging (block-wide, once) ----
  for (int i = tid; i < 256 * 64; i += blockDim.x) {
    int col = i >> 6, p = i & 63;
    smem[A_OFF + col * A_PITCH + p] = Ag[i];       // A[col(256)][p(64)]
  }
  for (int i = tid; i < 64 * 16; i += blockDim.x) {
    int h = i >> 4, k = i & 15;
    smem[W_OFF + h * W_PITCH + k] = Wg[i];         // W[h(64)][k(16)]
  }
  for (int i = tid; i < 64; i += blockDim.x) smem[B_OFF + i] = bg[i];
  __syncthreads();

  const float* alds = &smem[A_OFF];
  const float* wlds = &smem[W_OFF];
  const float* blds = &smem[B_OFF];
  float* abuf = &smem[WV_OFF + wave * PW_SIZE + PW_ABUF];
  float* mbuf = &smem[WV_OFF + wave * PW_SIZE + PW_MBUF];
  float* nbuf = &smem[WV_OFF + wave * PW_SIZE + PW_NBUF];

  // lane -> batch element (duplicated across wave halves)
  const int e  = l16;
  const int eg = blockIdx.x * 64 + wave * 16 + e;

  float x[16], v[16];
#pragma unroll
  for (int i = 0; i < 16; i++) {
    x[i] = X0[eg * 16 + i];
    v[i] = V0[eg * 16 + i];
  }

  float breg[4];
#pragma unroll
  for (int nc = 0; nc < 4; nc++) breg[nc] = blds[nc * 16 + l16];

  const long long PB = (long long)(steps + 1) * Btot * 16;  // velocity block

  if (lane < 16) {  // row 0: initial state
#pragma unroll
    for (int i = 0; i < 16; i++) {
      out[(long long)eg * 16 + i]      = x[i];
      out[PB + (long long)eg * 16 + i] = v[i];
    }
  }

  for (int s = 0; s < steps; s++) {
    float U[4][8];  // 1-h^2 in WMMA C-layout [nchunk][vgpr]

    // ================= Y = X*W^T + b ; H=tanh -> abuf ; U=1-h^2 ============
#pragma unroll
    for (int nc = 0; nc < 4; nc++) {
      v8f acc = (v8f)0.0f;
#pragma unroll
      for (int kc = 0; kc < 4; kc++) {
        v2f a, b;
        a.x = half ? x[4 * kc + 2] : x[4 * kc + 0];
        a.y = half ? x[4 * kc + 3] : x[4 * kc + 1];
        int n = nc * 16 + l16;                      // hidden index
        b.x = wlds[n * W_PITCH + 4 * kc + 2 * half + 0];
        b.y = wlds[n * W_PITCH + 4 * kc + 2 * half + 1];
        acc = wmma4(a, b, acc);
      }
#pragma unroll
      for (int r = 0; r < 8; r++) {
        float h = fast_tanh(acc[r] + breg[nc]);
        U[nc][r] = 1.0f - h * h;
        abuf[(r + 8 * half) * ABUF_PITCH + nc * 16 + l16] = h;  // H[el][hid]
      }
    }
    __syncthreads();

    // ================= Mten = H @ A^T -> mbuf[e][l(17)][k] =================
    for (int nc = 0; nc < 16; nc++) {
      v8f acc = (v8f)0.0f;
#pragma unroll
      for (int kc = 0; kc < 16; kc++) {
        v2f a, b;
        int p = 4 * kc + 2 * half;                  // hidden (K)
        a.x = abuf[l16 * ABUF_PITCH + p];
        a.y = abuf[l16 * ABUF_PITCH + p + 1];
        int col = nc * 16 + l16;                    // output col (N)
        b.x = alds[col * A_PITCH + p];
        b.y = alds[col * A_PITCH + p + 1];
        acc = wmma4(a, b, acc);
      }
#pragma unroll
      for (int r = 0; r < 8; r++)
        mbuf[(r + 8 * half) * M_EPITCH + nc * M_RPITCH + l16] = acc[r];
    }
    __syncthreads();

    // ================= Q = V*W^T ; S = U .* Q -> abuf ======================
#pragma unroll
    for (int nc = 0; nc < 4; nc++) {
      v8f acc = (v8f)0.0f;
#pragma unroll
      for (int kc = 0; kc < 4; kc++) {
        v2f a, b;
        a.x = half ? v[4 * kc + 2] : v[4 * kc + 0];
        a.y = half ? v[4 * kc + 3] : v[4 * kc + 1];
        int n = nc * 16 + l16;
        b.x = wlds[n * W_PITCH + 4 * kc + 2 * half + 0];
        b.y = wlds[n * W_PITCH + 4 * kc + 2 * half + 1];
        acc = wmma4(a, b, acc);
      }
#pragma unroll
      for (int r = 0; r < 8; r++)
        abuf[(r + 8 * half) * ABUF_PITCH + nc * 16 + l16] = acc[r] * U[nc][r];
    }
    __syncthreads();

    // ================= Nten = S @ A^T -> nbuf ==============================
    for (int nc = 0; nc < 16; nc++) {
      v8f acc = (v8f)0.0f;
#pragma unroll
      for (int kc = 0; kc < 16; kc++) {
        v2f a, b;
        int p = 4 * kc + 2 * half;
        a.x = abuf[l16 * ABUF_PITCH + p];
        a.y = abuf[l16 * ABUF_PITCH + p + 1];
        int col = nc * 16 + l16;
        b.x = alds[col * A_PITCH + p];
        b.y = alds[col * A_PITCH + p + 1];
        acc = wmma4(a, b, acc);
      }
#pragma unroll
      for (int r = 0; r < 8; r++)
        nbuf[(r + 8 * half) * M_EPITCH + nc * M_RPITCH + l16] = acc[r];
    }
    __syncthreads();

    // ============ T1_l = 0.5 * sum_k (N[l][k] + N[k][l]) v_k ===============
    float T1[16];
#pragma unroll
    for (int l = 0; l < 16; l++) {
      float ssum = 0.0f;
#pragma unroll
      for (int k = 0; k < 16; k++)
        ssum += (nbuf[e * M_EPITCH + l * M_RPITCH + k] +
                 nbuf[e * M_EPITCH + k * M_RPITCH + l]) * v[k];
      T1[l] = 0.5f * ssum;
    }

    // ============ R = vec(v v^T) @ A ; UR -> abuf  (K = 256) ===============
#pragma unroll
    for (int ncr = 0; ncr < 4; ncr++) {
      v8f acc = (v8f)0.0f;
#pragma unroll
      for (int kc = 0; kc < 64; kc++) {            // vv generated on the fly
        int k0 = 4 * kc;
        float a0 = v[(k0 + 0) >> 4] * v[(k0 + 0) & 15];
        float a1 = v[(k0 + 1) >> 4] * v[(k0 + 1) & 15];
        float a2 = v[(k0 + 2) >> 4] * v[(k0 + 2) & 15];
        float a3 = v[(k0 + 3) >> 4] * v[(k0 + 3) & 15];
        v2f a, b;
        a.x = half ? a2 : a0;
        a.y = half ? a3 : a1;
        int colk = k0 + 2 * half;                  // K = flattened (j,k)
        int n = ncr * 16 + l16;                    // hidden (N)
        b.x = alds[(colk + 0) * A_PITCH + n];
        b.y = alds[(colk + 1) * A_PITCH + n];
        acc = wmma4(a, b, acc);
      }
#pragma unroll
      for (int r = 0; r < 8; r++)
        abuf[(r + 8 * half) * ABUF_PITCH + ncr * 16 + l16] = acc[r] * U[ncr][r];
    }
    __syncthreads();

    // ================= T3 = (U.*R) @ W  (K=64, N=16) =======================
    v8f t3acc = (v8f)0.0f;
#pragma unroll
    for (int kc = 0; kc < 16; kc++) {
      v2f a, b;
      int p = 4 * kc + 2 * half;
      a.x = abuf[l16 * ABUF_PITCH + p];
      a.y = abuf[l16 * ABUF_PITCH + p + 1];
      b.x = wlds[(p + 0) * W_PITCH + l16];         // W[p][l]
      b.y = wlds[(p + 1) * W_PITCH + l16];
      t3acc = wmma4(a, b, t3acc);
    }
    __syncthreads();
#pragma unroll
    for (int r = 0; r < 8; r++)                    // transpose T3 via LDS
      abuf[(r + 8 * half) * ABUF_PITCH + l16] = t3acc[r];
    __syncthreads();

    float cv[16];
#pragma unroll
    for (int l = 0; l < 16; l++)
      cv[l] = T1[l] - 0.5f * abuf[e * ABUF_PITCH + l];

    // ============ E = 0.5(M + M^T) -> nbuf[e][i*16+l] (fp32, v4-aligned) ===
    if (lane < 16) {
#pragma unroll
      for (int i = 0; i < 16; i++)
#pragma unroll
        for (int l = 0; l < 16; l++)
          nbuf[e * M_EPITCH + i * 16 + l] =
              0.5f * (mbuf[e * M_EPITCH + i * M_RPITCH + l] +
                      mbuf[e * M_EPITCH + l * M_RPITCH + i]);
    }
    __syncthreads();

    // ============ Neumann solve (I+E) a = c : z <- c - E z ================
    float z[16];
#pragma unroll
    for (int i = 0; i < 16; i++) z[i] = cv[i];
    for (int it = 0; it < NEUM_ITERS; it++) {
      float t[16];
#pragma unroll
      for (int i = 0; i < 16; i++) {
        const v4f* row = (const v4f*)(nbuf + e * M_EPITCH + i * 16);
        v4f r0 = row[0], r1 = row[1], r2 = row[2], r3 = row[3];
        float acc = r0.x * z[0] + r0.y * z[1] + r0.z * z[2] + r0.w * z[3];
        acc += r1.x * z[4] + r1.y * z[5] + r1.z * z[6] + r1.w * z[7];
        acc += r2.x * z[8] + r2.y * z[9] + r2.z * z[10] + r2.w * z[11];
        acc += r3.x * z[12] + r3.y * z[13] + r3.z * z[14] + r3.w * z[15];
        t[i] = cv[i] - acc;
      }
#pragma unroll
      for (int i = 0; i < 16; i++) z[i] = t[i];
    }

    // ================= update & store =====================================
#pragma unroll
    for (int i = 0; i < 16; i++) {
      v[i] -= STEP_SIZE * z[i];   // v_upd = -Gamma v v = -z
      x[i] += STEP_SIZE * v[i];
    }
    if (lane < 16) {
      long long base = (long long)(s + 1) * Btot * 16 + (long long)eg * 16;
#pragma unroll
      for (int i = 0; i < 16; i++) {
        out[base + i]      = x[i];
        out[PB + base + i] = v[i];
      }
    }
    __syncthreads();  // protect mbuf/nbuf/abuf reuse next step
  }
}

extern "C" void kernel_launch(void* const* d_in, const int* in_sizes, int n_in,
                              void* d_out, int out_size, void* d_ws, size_t ws_size,
                              hipStream_t stream) {
  const float* X0 = (const float*)d_in[0];   // [8192,16]
  const float* V0 = (const float*)d_in[1];   // [8192,16]
  const float* Wg = (const float*)d_in[2];   // [64,16]
  const float* bg = (const float*)d_in[3];   // [64]
  const float* Ag = (const float*)d_in[4];   // [256,64]
  const int* steps = (const int*)d_in[5];    // scalar
  float* out = (float*)d_out;

  int Btot = in_sizes[0] / 16;               // 8192
  int blocks = Btot / 64;                    // 4 waves x 16 elems per block
  size_t shmem = (size_t)LDS_FLOATS * sizeof(float);  // ~233 KB (<320 KB/WGP)

  (void)hipFuncSetAttribute((const void*)geodesic_wmma_kernel,
                            hipFuncAttributeMaxDynamicSharedMemorySize,
                            (int)shmem);

  geodesic_wmma_kernel<<<blocks, 128, shmem, stream>>>(X0, V0, Wg, bg, Ag,
                                                       steps, out, Btot);
  (void)d_ws; (void)ws_size; (void)n_in; (void)out_size;
}